// CrossAttention_455266534011
// MI455X (gfx1250) — compile-verified
//
#include <hip/hip_runtime.h>
#include <hip/hip_bf16.h>

// Problem sizes (fixed by the reference)
#define BT    8
#define CD    256
#define Hh    64
#define Ww    64
#define HEADS 8
#define NL    256      // (H/4)*(W/4)
#define NH    4096     // H*W

typedef __attribute__((ext_vector_type(2))) float v2f;
typedef __attribute__((ext_vector_type(8))) float v8f;

// ---------------------------------------------------------------------------
// Pass 1: sel[b,i] = argmax_j sum_h attn[b,h,i,j]   (mean == sum for argmax)
// One block per (b,i); 256 threads, thread j owns column j.
// ---------------------------------------------------------------------------
__global__ void k_argmax(const float* __restrict__ attn, int* __restrict__ sel) {
    const int bi = blockIdx.x;            // 0..2047
    const int b  = bi >> 8;
    const int i  = bi & 255;
    const int j  = threadIdx.x;

    const float* p = attn + ((((size_t)b * HEADS) * NL + i) * NL) + j;
    float s = 0.f;
#pragma unroll
    for (int h = 0; h < HEADS; ++h)
        s += p[(size_t)h * NL * NL];

    __shared__ float sv[256];
    __shared__ int   si[256];
    sv[j] = s; si[j] = j;
    __syncthreads();
#pragma unroll
    for (int off = 128; off > 0; off >>= 1) {
        if (j < off) {
            float a = sv[j], c = sv[j + off];
            int   ib = si[j + off];
            // strictly greater wins; on tie keep the lower index (top_k rule)
            if (c > a || (c == a && ib < si[j])) { sv[j] = c; si[j] = ib; }
        }
        __syncthreads();
    }
    if (j == 0) sel[bi] = si[0];
}

// ---------------------------------------------------------------------------
// Pass 2: bm[b, blk, c] = (1/16) * sum of C[b,c,:,:] over 4x4 block `blk`
// computed as D = A x P with V_WMMA_F32_16X16X4_F32.
//   One wave per (b, channel-tile t, block-row r): D is 16 channels x 16 blocks.
//   A (16x4 f32, documented layout): lanes 0-15 hold K=0,1 / lanes 16-31 K=2,3,
//     M = lane&15 = local channel.  A[m,k] = C[b, 16t+m, 4r+dh, 4g+k].
//   B (4x16): P[k,n] = (n==g) ? 1/16 : 0  -- k-independent, so both B VGPRs on
//     every lane hold ((lane&15)==g)/16, which is layout-robust.
// 64 accumulating WMMAs per wave cover dh=0..3, g=0..15.
// ---------------------------------------------------------------------------
__global__ void k_blockmean(const float* __restrict__ Cin, float* __restrict__ bm) {
    const int wave = (blockIdx.x * blockDim.x + threadIdx.x) >> 5;  // 0..2047
    const int lane = threadIdx.x & 31;
    const int b    = wave >> 8;
    const int rem  = wave & 255;
    const int t    = rem >> 4;        // channel tile 0..15
    const int r    = rem & 15;        // block row    0..15
    const int cl   = lane & 15;       // local channel (A rows) / block col (B,D cols)
    const int koff = (lane < 16) ? 0 : 2;

    const float* base = Cin + (((size_t)b * CD + (t * 16 + cl)) << 12); // *4096

    v8f acc = {0.f, 0.f, 0.f, 0.f, 0.f, 0.f, 0.f, 0.f};
    for (int dh = 0; dh < 4; ++dh) {
        const float* rowp = base + (((r << 2) + dh) << 6) + koff;
#pragma unroll
        for (int g = 0; g < 16; ++g) {
            v2f a = *(const v2f*)(rowp + (g << 2));       // 8B aligned
            float bv = (cl == g) ? 0.0625f : 0.0f;
            v2f bb = {bv, bv};
            acc = __builtin_amdgcn_wmma_f32_16x16x4_f32(
                /*neg_a=*/false, a, /*neg_b=*/false, bb,
                /*c_mod=*/(short)0, acc, /*reuse_a=*/false, /*reuse_b=*/false);
        }
    }

    // D layout: VGPR v -> M = v (lanes 0-15) / v+8 (lanes 16-31); N = lane&15
    const int mh  = (lane < 16) ? 0 : 8;
    const int blk = (r << 4) + cl;
#pragma unroll
    for (int v = 0; v < 8; ++v) {
        const int c = t * 16 + v + mh;
        bm[(((size_t)b * NL + blk) << 8) + c] = acc[v];
    }
}

// ---------------------------------------------------------------------------
// Pass 3: out[b,c,q] = bm[b, sel[b, cell(q)], c]
// One block per (b,c). Stage the 256 gathered means in LDS, then emit fully
// coalesced writes of the 16KB output image.
// ---------------------------------------------------------------------------
__global__ void k_scatter(const float* __restrict__ bm, const int* __restrict__ sel,
                          float* __restrict__ out) {
    const int bc = blockIdx.x;            // 0..2047
    const int b  = bc >> 8;
    const int c  = bc & 255;
    const int tid = threadIdx.x;

    __shared__ float vals[NL];
    const int s = sel[(b << 8) + tid];
    vals[tid] = bm[(((size_t)b * NL + s) << 8) + c];
    __syncthreads();

    float* o = out + (((size_t)b * CD + c) << 12);
#pragma unroll
    for (int l = 0; l < 16; ++l) {
        const int q = (l << 8) + tid;
        const int cell = ((q >> 8) << 4) + ((q & 63) >> 2);   // (h/4)*16 + (w/4); h = q>>6
        o[q] = vals[cell];
    }
}

// ---------------------------------------------------------------------------
extern "C" void kernel_launch(void* const* d_in, const int* in_sizes, int n_in,
                              void* d_out, int out_size, void* d_ws, size_t ws_size,
                              hipStream_t stream) {
    // inputs: A, B, C, D, attn  (all float32); only C and attn are used.
    const float* Cin  = (const float*)d_in[2];
    const float* attn = (const float*)d_in[4];
    float* out = (float*)d_out;

    // workspace: [0, 8KB)   sel  (2048 int)
    //            [8KB, +2MB) bm  (8*256*256 float)
    int*   sel = (int*)d_ws;
    float* bm  = (float*)((char*)d_ws + 8192);

    k_argmax   <<<BT * NL, 256, 0, stream>>>(attn, sel);
    k_blockmean<<<(BT * 16 * 16) / 8, 256, 0, stream>>>(Cin, bm);
    k_scatter  <<<BT * CD, 256, 0, stream>>>(bm, sel, out);
}